// CustomMultiHeadAttention_14448269984057
// MI455X (gfx1250) — compile-verified
//
#include <hip/hip_runtime.h>

#define BB 32
#define SS 1024
#define DD 256
#define HH 8

typedef __attribute__((ext_vector_type(2))) float v2f;
typedef __attribute__((ext_vector_type(8))) float v8f;

// workspace layout (float offsets)
#define OFF_QK 0                               // H*D = 2048
#define OFF_CB 2048                            // H   = 8
#define OFF_SC 4096                            // B*H*S = 262144  (scores/attn, [b][h][s])
#define OFF_XA (OFF_SC + BB * HH * SS)         // B*H*D = 65536   ([b][h][d])
#define OFF_PL (OFF_XA + BB * HH * DD)         // B*H*D = 65536   ([b][h*D+d])
// total = 397312 floats = ~1.6 MB of d_ws

// ---------------------------------------------------------------------------
// K1: fold query into Wk:  qk[h][i] = sum_d Wk[i, h*D+d] * query[h][d]
//     cb[h] = bk_h . query[h]
__global__ void k1_fold_query(const float* __restrict__ Wk,
                              const float* __restrict__ bk,
                              const float* __restrict__ query,
                              float* __restrict__ ws) {
    int t = blockIdx.x * blockDim.x + threadIdx.x;   // 0..2047
    int h = t >> 8, i = t & 255;
    const float* q    = query + h * DD;
    const float* wrow = Wk + (size_t)i * (DD * HH) + h * DD;
    float s = 0.f;
#pragma unroll 8
    for (int d = 0; d < DD; ++d) s += wrow[d] * q[d];
    ws[OFF_QK + t] = s;
    if (i == 0) {
        const float* bh = bk + h * DD;
        float cb = 0.f;
        for (int d = 0; d < DD; ++d) cb += bh[d] * q[d];
        ws[OFF_CB + h] = cb;
    }
}

// ---------------------------------------------------------------------------
// K2: scores[b][h][s] = x[row,:] . qk[h] + cb[h]   via V_WMMA_F32_16X16X4_F32
//     X: [32768 x 256], B-matrix: qk^T zero-padded to N=16 *in LDS* so the
//     K-loop has no per-iteration select ops. 128 thr = 4 waves, each wave
//     owns a 16-row tile; K-loop of 64 x (K=4) WMMAs.
__global__ void k2_scores(const float* __restrict__ x, float* __restrict__ ws) {
    __shared__ float qs[16 * DD];   // 16 KB: rows 0-7 = qk, rows 8-15 = 0
    __shared__ float cbs[16];
    int tid = threadIdx.x;          // 128
    for (int i = tid; i < 16 * DD; i += 128)
        qs[i] = (i < HH * DD) ? ws[OFF_QK + i] : 0.f;
    if (tid < 16) cbs[tid] = (tid < HH) ? ws[OFF_CB + tid] : 0.f;
    __syncthreads();

    int wave = tid >> 5, lane = tid & 31;
    int r0    = (blockIdx.x * 4 + wave) * 16;   // 16-row tile base
    int mn    = lane & 15;                      // M for A, N for B/D
    int khalf = lane >> 4;                      // lanes 16-31 carry K=2,3

    float cbv = cbs[mn];                        // bias folded into C init
    v8f acc = {cbv, cbv, cbv, cbv, cbv, cbv, cbv, cbv};

    const float* arow = x + (size_t)(r0 + mn) * DD + 2 * khalf;
    const float* brow = qs + mn * DD + 2 * khalf;

    for (int kk = 0; kk < DD; kk += 4) {
        __builtin_prefetch(arow + kk + 128, 0, 3);     // global_prefetch_b8 (WGP)
        float2 av = *(const float2*)(arow + kk);       // A[m][kk+2kh .. +1]
        float2 bv = *(const float2*)(brow + kk);       // B[k][n] = qk[n][k]
        v2f a;  a.x = av.x;  a.y = av.y;
        v2f bf; bf.x = bv.x; bf.y = bv.y;
        acc = __builtin_amdgcn_wmma_f32_16x16x4_f32(
                  false, a, false, bf, (short)0, acc, false, false);
    }
    // D layout: lanes 0-15 -> M = v, lanes 16-31 -> M = v+8; N = mn
    if (mn < HH) {
#pragma unroll
        for (int v = 0; v < 8; ++v) {
            int grow = r0 + v + 8 * khalf;
            int b = grow >> 10, s = grow & 1023;
            ws[OFF_SC + (size_t)(b * HH + mn) * SS + s] = acc[v];
        }
    }
}

// ---------------------------------------------------------------------------
// K3: softmax over S=1024 for each (b,h); one 256-thread block per row.
__global__ void k3_softmax(float* __restrict__ ws) {
    __shared__ float red[256];
    int tid = threadIdx.x;
    float* p = ws + OFF_SC + (size_t)blockIdx.x * SS;
    float v0 = p[tid], v1 = p[tid + 256], v2 = p[tid + 512], v3 = p[tid + 768];
    float m = fmaxf(fmaxf(v0, v1), fmaxf(v2, v3));
    red[tid] = m; __syncthreads();
    for (int off = 128; off > 0; off >>= 1) {
        if (tid < off) red[tid] = fmaxf(red[tid], red[tid + off]);
        __syncthreads();
    }
    m = red[0]; __syncthreads();
    v0 = __expf(v0 - m); v1 = __expf(v1 - m);
    v2 = __expf(v2 - m); v3 = __expf(v3 - m);
    red[tid] = v0 + v1 + v2 + v3; __syncthreads();
    for (int off = 128; off > 0; off >>= 1) {
        if (tid < off) red[tid] += red[tid + off];
        __syncthreads();
    }
    float inv = 1.0f / red[0];
    p[tid] = v0 * inv; p[tid + 256] = v1 * inv;
    p[tid + 512] = v2 * inv; p[tid + 768] = v3 * inv;
}

// ---------------------------------------------------------------------------
// K4: xa[b][h][:] = attn[b] (8x1024, zero-padded to M=16 in LDS) @ x[b]
//     (1024x256), via WMMA f32 K=4. One block per batch; padded attn tile in
//     64 KB LDS (WGP has 320 KB) so the K-loop is select-free. 8 waves x 2
//     N-tiles cover D=256. x[b] is L2-resident from K2 (33.5 MB << 192 MB L2).
__global__ void k4_pool(const float* __restrict__ x, float* __restrict__ ws) {
    __shared__ float at[16 * SS];   // 64 KB: rows 0-7 = attn, rows 8-15 = 0
    int tid = threadIdx.x;          // 256
    int b = blockIdx.x;
    const float* attn = ws + OFF_SC + (size_t)b * HH * SS;
    for (int i = tid; i < 16 * SS; i += 256)
        at[i] = (i < HH * SS) ? attn[i] : 0.f;
    __syncthreads();

    int wave = tid >> 5, lane = tid & 31;
    int mn = lane & 15, khalf = lane >> 4;
    const float* xb = x + (size_t)b * SS * DD;
    const float* arow = at + mn * SS + 2 * khalf;

    for (int t = 0; t < 2; ++t) {
        int nt  = wave * 2 + t;
        int col = nt * 16 + mn;
        v8f acc = {0.f, 0.f, 0.f, 0.f, 0.f, 0.f, 0.f, 0.f};
        for (int kk = 0; kk < SS; kk += 4) {
            int krow = kk + 2 * khalf;
            float2 av = *(const float2*)(arow + kk);   // A[m][kk+2kh .. +1]
            v2f a;  a.x = av.x;  a.y = av.y;
            v2f bf;                                    // B[k][n] = x[b][k][col]
            bf.x = xb[(size_t)krow * DD + col];
            bf.y = xb[(size_t)(krow + 1) * DD + col];
            acc = __builtin_amdgcn_wmma_f32_16x16x4_f32(
                      false, a, false, bf, (short)0, acc, false, false);
        }
        if (khalf == 0) {                        // lanes 0-15 hold M=0..7 (heads)
#pragma unroll
            for (int v = 0; v < 8; ++v)
                ws[OFF_XA + (size_t)(b * HH + v) * DD + col] = acc[v];
        }
    }
}

// ---------------------------------------------------------------------------
// K5: pooled[b][j] = xa[b][j>>8][:] . Wv[:, j] + bv[j]   (Wv column-coalesced)
__global__ void k5_project_v(const float* __restrict__ Wv,
                             const float* __restrict__ bv,
                             float* __restrict__ ws) {
    int b = blockIdx.x >> 3;
    int j = ((blockIdx.x & 7) << 8) + threadIdx.x;   // 0..2047
    int h = j >> 8;
    const float* xa = ws + OFF_XA + (size_t)(b * HH + h) * DD;
    float acc = bv[j];
#pragma unroll 4
    for (int i = 0; i < DD; ++i) acc += xa[i] * Wv[(size_t)i * (DD * HH) + j];
    ws[OFF_PL + (size_t)b * (DD * HH) + j] = acc;
}

// ---------------------------------------------------------------------------
// K6: out[b][c] = pooled[b][:] . Wo[:, c] + bo[c]
__global__ void k6_out(const float* __restrict__ Wo,
                       const float* __restrict__ bo,
                       const float* __restrict__ ws,
                       float* __restrict__ out) {
    int b = blockIdx.x, c = threadIdx.x;
    const float* pl = ws + OFF_PL + (size_t)b * (DD * HH);
    float acc = bo[c];
#pragma unroll 4
    for (int j = 0; j < DD * HH; ++j) acc += pl[j] * Wo[(size_t)j * DD + c];
    out[b * DD + c] = acc;
}

// ---------------------------------------------------------------------------
extern "C" void kernel_launch(void* const* d_in, const int* in_sizes, int n_in,
                              void* d_out, int out_size, void* d_ws, size_t ws_size,
                              hipStream_t stream) {
    const float* x  = (const float*)d_in[0];
    const float* Wk = (const float*)d_in[1];
    const float* bk = (const float*)d_in[2];
    const float* Wv = (const float*)d_in[3];
    const float* bv = (const float*)d_in[4];
    const float* q  = (const float*)d_in[5];
    const float* Wo = (const float*)d_in[6];
    const float* bo = (const float*)d_in[7];
    float* ws  = (float*)d_ws;
    float* out = (float*)d_out;

    k1_fold_query<<<8, 256, 0, stream>>>(Wk, bk, q, ws);
    k2_scores    <<<512, 128, 0, stream>>>(x, ws);     // 32768 rows / (4 waves * 16)
    k3_softmax   <<<BB * HH, 256, 0, stream>>>(ws);
    k4_pool      <<<BB, 256, 0, stream>>>(x, ws);
    k5_project_v <<<256, 256, 0, stream>>>(Wv, bv, ws);
    k6_out       <<<BB, 256, 0, stream>>>(Wo, bo, ws, out);
}